// MinCutNet_30279519436810
// MI455X (gfx1250) — compile-verified
//
#include <hip/hip_runtime.h>
#include <math.h>

typedef __attribute__((ext_vector_type(2))) float v2f;
typedef __attribute__((ext_vector_type(8))) float v8f;

#define TOTAL 65536
#define NGRAPH 64
#define NN 1024
#define NE 2097152
#define INC 128
#define HID 32
#define KCL 16
#define OUTC 10

// workspace offsets (in floats)
#define OFF_H0  0u
#define OFF_AGG 2097152u
#define OFF_H1  4194304u
#define OFF_S   6291456u
#define OFF_AS  7340032u
#define OFF_DEG 8388608u
#define OFF_OUT 8454144u   // [64,16,32]
#define OFF_OA  8486912u   // [64,16,16]
#define OFF_SS  8503296u   // [64,16,16]
#define OFF_DEN 8519680u   // [64]
#define WS_END  8519744u

static __device__ __forceinline__ v8f wmma4(v2f a, v2f b, v8f c) {
  return __builtin_amdgcn_wmma_f32_16x16x4_f32(false, a, false, b, (short)0, c,
                                               false, false);
}

__global__ void k_zero(float* __restrict__ p, unsigned n) {
  unsigned i = blockIdx.x * blockDim.x + threadIdx.x;
  unsigned stride = gridDim.x * blockDim.x;
  for (; i < n; i += stride) p[i] = 0.0f;
}

// h0 = x @ W1^T + b1   — f32 WMMA 16x16x4, one wave per 16 nodes
__global__ void k_lin1(const float* __restrict__ x, const float* __restrict__ W1,
                       const float* __restrict__ b1, float* __restrict__ h0) {
  const int tid = blockIdx.x * blockDim.x + threadIdx.x;
  const int wave = tid >> 5;
  const int lane = tid & 31;
  const int lh = lane & 15;
  const int hi = lane >> 4;           // 0 | 1
  const int rowb = wave * 16;         // node base
  v8f c0 = {}; v8f c1 = {};
  for (int k0 = 0; k0 < INC; k0 += 4) {
    // A tile: x[rowb+lh][k0+2*hi], x[..][k0+2*hi+1]
    const float* ap = x + (size_t)(rowb + lh) * INC + k0 + 2 * hi;
    v2f a; a.x = ap[0]; a.y = ap[1];
    // B tiles: W1^T[k][n] = W1[n*INC + k], n-halves 0..15 and 16..31
    const float* bp0 = W1 + (size_t)(lh)      * INC + k0 + 2 * hi;
    const float* bp1 = W1 + (size_t)(16 + lh) * INC + k0 + 2 * hi;
    v2f b0; b0.x = bp0[0]; b0.y = bp0[1];
    v2f b1v; b1v.x = bp1[0]; b1v.y = bp1[1];
    c0 = wmma4(a, b0, c0);
    c1 = wmma4(a, b1v, c1);
  }
  const float bia0 = b1[lh], bia1 = b1[16 + lh];
#pragma unroll
  for (int v = 0; v < 8; ++v) {
    int row = rowb + v + 8 * hi;
    h0[(size_t)row * HID + lh]      = c0[v] + bia0;
    h0[(size_t)row * HID + 16 + lh] = c1[v] + bia1;
  }
}

// edge pass 1: agg[src] += h0[dst] ; deg[src] += 1   (8 threads / edge)
__global__ void k_edge1(const int* __restrict__ ei, const float* __restrict__ h0,
                        float* __restrict__ agg, float* __restrict__ deg) {
  unsigned gid = blockIdx.x * blockDim.x + threadIdx.x;
  unsigned e = gid >> 3;
  if (e >= NE) return;
  int c = (gid & 7) * 4;
  int src = ei[e];
  int dst = ei[NE + e];
  const float4 v = *(const float4*)(h0 + (size_t)dst * HID + c);
  float* ap = agg + (size_t)src * HID + c;
  atomicAdd(ap + 0, v.x); atomicAdd(ap + 1, v.y);
  atomicAdd(ap + 2, v.z); atomicAdd(ap + 3, v.w);
  if ((gid & 7) == 0) atomicAdd(deg + src, 1.0f);
}

// per-node: h1 = agg@Wrel1^T + brel1 + h0@Wroot1^T ; s = softmax(h1@Wp^T+bp)
// also den[b] += deg[n] * sum_k s[n,k]^2
__global__ void k_node(const float* __restrict__ agg, const float* __restrict__ h0,
                       const float* __restrict__ Wrel1, const float* __restrict__ brel1,
                       const float* __restrict__ Wroot1,
                       const float* __restrict__ Wp, const float* __restrict__ bp,
                       const float* __restrict__ deg,
                       float* __restrict__ h1, float* __restrict__ s,
                       float* __restrict__ den) {
  __shared__ float sWrel[HID * HID], sWroot[HID * HID], sbrel[HID];
  __shared__ float sWp[KCL * HID], sbp[KCL];
  __shared__ float red[256];
  const int tid = threadIdx.x;
  for (int i = tid; i < HID * HID; i += 256) { sWrel[i] = Wrel1[i]; sWroot[i] = Wroot1[i]; }
  for (int i = tid; i < KCL * HID; i += 256) sWp[i] = Wp[i];
  if (tid < HID) sbrel[tid] = brel1[tid];
  if (tid < KCL) sbp[tid] = bp[tid];
  __syncthreads();

  const int node = blockIdx.x * 256 + tid;
  float a[HID], r[HID], h[HID];
#pragma unroll
  for (int i = 0; i < HID; ++i) { a[i] = agg[(size_t)node * HID + i]; r[i] = h0[(size_t)node * HID + i]; }
  for (int o = 0; o < HID; ++o) {
    float acc = sbrel[o];
#pragma unroll
    for (int i = 0; i < HID; ++i) acc += sWrel[o * HID + i] * a[i] + sWroot[o * HID + i] * r[i];
    h[o] = acc;
    h1[(size_t)node * HID + o] = acc;
  }
  float lg[KCL];
  float mx = -1e30f;
  for (int k = 0; k < KCL; ++k) {
    float acc = sbp[k];
#pragma unroll
    for (int i = 0; i < HID; ++i) acc += sWp[k * HID + i] * h[i];
    lg[k] = acc; mx = fmaxf(mx, acc);
  }
  float sum = 0.0f;
#pragma unroll
  for (int k = 0; k < KCL; ++k) { lg[k] = expf(lg[k] - mx); sum += lg[k]; }
  const float inv = 1.0f / sum;
  float sq = 0.0f;
#pragma unroll
  for (int k = 0; k < KCL; ++k) {
    float sv = lg[k] * inv;
    s[(size_t)node * KCL + k] = sv;
    sq += sv * sv;
  }
  red[tid] = deg[node] * sq;
  __syncthreads();
  for (int off = 128; off > 0; off >>= 1) {
    if (tid < off) red[tid] += red[tid + off];
    __syncthreads();
  }
  if (tid == 0) atomicAdd(den + (blockIdx.x >> 2), red[0]);  // 4 blocks / graph
}

// edge pass 2: as[src] += s[dst]   (4 threads / edge)
__global__ void k_edge2(const int* __restrict__ ei, const float* __restrict__ s,
                        float* __restrict__ as_) {
  unsigned gid = blockIdx.x * blockDim.x + threadIdx.x;
  unsigned e = gid >> 2;
  if (e >= NE) return;
  int c = (gid & 3) * 4;
  int src = ei[e];
  int dst = ei[NE + e];
  const float4 v = *(const float4*)(s + (size_t)dst * KCL + c);
  float* p = as_ + (size_t)src * KCL + c;
  atomicAdd(p + 0, v.x); atomicAdd(p + 1, v.y);
  atomicAdd(p + 2, v.z); atomicAdd(p + 3, v.w);
}

// per-graph pooled reductions via WMMA: out=S^T H, oa=S^T(AS), ss=S^T S
// 4 waves per graph, each reduces 256 nodes (K chunks), atomic flush
__global__ void k_pool(const float* __restrict__ s, const float* __restrict__ h1,
                       const float* __restrict__ as_,
                       float* __restrict__ outS, float* __restrict__ oa,
                       float* __restrict__ ssb) {
  const int tid = blockIdx.x * blockDim.x + threadIdx.x;
  const int wave = tid >> 5;            // 0..255
  const int lane = tid & 31;
  const int b = wave >> 2;
  const int chunk = wave & 3;
  const int lh = lane & 15, hi = lane >> 4;
  const int gb = b * NN + chunk * 256;
  v8f cO0 = {}, cO1 = {}, cA = {}, cS = {};
  for (int it = 0; it < 64; ++it) {
    const int r0 = gb + it * 4 + 2 * hi;          // node for .x (r0+1 for .y)
    // A tile of S^T and B tile of S coincide element-for-element:
    v2f a;   a.x  = s[(size_t)r0 * KCL + lh];        a.y  = s[(size_t)(r0 + 1) * KCL + lh];
    v2f bh0; bh0.x = h1[(size_t)r0 * HID + lh];      bh0.y = h1[(size_t)(r0 + 1) * HID + lh];
    v2f bh1; bh1.x = h1[(size_t)r0 * HID + 16 + lh]; bh1.y = h1[(size_t)(r0 + 1) * HID + 16 + lh];
    v2f ba;  ba.x = as_[(size_t)r0 * KCL + lh];      ba.y = as_[(size_t)(r0 + 1) * KCL + lh];
    cO0 = wmma4(a, bh0, cO0);
    cO1 = wmma4(a, bh1, cO1);
    cA  = wmma4(a, ba, cA);
    cS  = wmma4(a, a, cS);
  }
#pragma unroll
  for (int v = 0; v < 8; ++v) {
    const int m = v + 8 * hi;
    atomicAdd(outS + (size_t)b * KCL * HID + m * HID + lh,      cO0[v]);
    atomicAdd(outS + (size_t)b * KCL * HID + m * HID + 16 + lh, cO1[v]);
    atomicAdd(oa  + (size_t)b * KCL * KCL + m * KCL + lh,       cA[v]);
    atomicAdd(ssb + (size_t)b * KCL * KCL + m * KCL + lh,       cS[v]);
  }
}

static __device__ __forceinline__ float blk_reduce(float v, float* red, int tid) {
  red[tid] = v; __syncthreads();
  for (int off = 32; off > 0; off >>= 1) {
    if (tid < off) red[tid] += red[tid + off];
    __syncthreads();
  }
  float r = red[0]; __syncthreads();
  return r;
}

// one block per graph (64 threads): losses + pooled conv + readout + heads
__global__ void k_final(const float* __restrict__ outS, const float* __restrict__ oa,
                        const float* __restrict__ ssb, const float* __restrict__ den,
                        const float* __restrict__ Wrel2, const float* __restrict__ brel2,
                        const float* __restrict__ Wroot2,
                        const float* __restrict__ W2, const float* __restrict__ b2,
                        const float* __restrict__ W3, const float* __restrict__ b3,
                        float* __restrict__ out) {
  const int b = blockIdx.x;
  const int tid = threadIdx.x;  // 64
  __shared__ float sOa[256], sOut[512], sT1[512], sH2[512];
  __shared__ float sD[16], sG[32], sG2[32], sG3[10], red[64];

  for (int i = tid; i < 256; i += 64) sOa[i] = oa[b * 256 + i];
  for (int i = tid; i < 512; i += 64) sOut[i] = outS[b * 512 + i];
  __syncthreads();

  // mincut numerator = trace(S^T A S)
  float num = blk_reduce(tid < 16 ? sOa[tid * 16 + tid] : 0.0f, red, tid);

  // ortho loss
  float ssq = 0.0f;
  for (int i = tid; i < 256; i += 64) { float v = ssb[b * 256 + i]; ssq += v * v; }
  const float ssnorm = sqrtf(blk_reduce(ssq, red, tid));
  float ol = 0.0f;
  for (int i = tid; i < 256; i += 64) {
    float v = ssb[b * 256 + i] / ssnorm;
    if ((i / 16) == (i % 16)) v -= 0.25f;  // I/sqrt(K), K=16
    ol += v * v;
  }
  const float ortho = sqrtf(blk_reduce(ol, red, tid));

  // zero diagonal + symmetric degree normalization
  if (tid < 16) sOa[tid * 16 + tid] = 0.0f;
  __syncthreads();
  if (tid < 16) {
    float rs = 0.0f;
    for (int l = 0; l < 16; ++l) rs += sOa[tid * 16 + l];
    sD[tid] = sqrtf(rs) + 1e-15f;
  }
  __syncthreads();
  for (int i = tid; i < 256; i += 64) sOa[i] = sOa[i] / (sD[i / 16] * sD[i % 16]);
  __syncthreads();

  // t1 = oa_n @ out  [16,32]
  for (int i = tid; i < 512; i += 64) {
    int k = i / 32, h = i % 32;
    float acc = 0.0f;
    for (int l = 0; l < 16; ++l) acc += sOa[k * 16 + l] * sOut[l * 32 + h];
    sT1[i] = acc;
  }
  __syncthreads();
  // h2 = t1 @ Wrel2^T + brel2 + out @ Wroot2^T
  for (int i = tid; i < 512; i += 64) {
    int k = i / 32, h = i % 32;
    float acc = brel2[h];
    for (int j = 0; j < 32; ++j)
      acc += Wrel2[h * 32 + j] * sT1[k * 32 + j] + Wroot2[h * 32 + j] * sOut[k * 32 + j];
    sH2[i] = acc;
  }
  __syncthreads();
  if (tid < 32) {
    float acc = 0.0f;
    for (int k = 0; k < 16; ++k) acc += sH2[k * 32 + tid];
    sG[tid] = acc;
  }
  __syncthreads();
  if (tid < 32) {
    float acc = b2[tid];
    for (int j = 0; j < 32; ++j) acc += W2[tid * 32 + j] * sG[j];
    sG2[tid] = fmaxf(acc, 0.0f);
  }
  __syncthreads();
  if (tid < 10) {
    float acc = b3[tid];
    for (int j = 0; j < 32; ++j) acc += W3[tid * 32 + j] * sG2[j];
    sG3[tid] = acc;
  }
  __syncthreads();
  if (tid == 0) {
    float mx = -1e30f;
    for (int o = 0; o < 10; ++o) mx = fmaxf(mx, sG3[o]);
    float se = 0.0f;
    for (int o = 0; o < 10; ++o) se += expf(sG3[o] - mx);
    float lse = mx + logf(se);
    for (int o = 0; o < 10; ++o) out[b * 10 + o] = sG3[o] - lse;
    atomicAdd(out + 640, -(num / den[b]) * (1.0f / 64.0f));
    atomicAdd(out + 641, ortho * (1.0f / 64.0f));
  }
}

extern "C" void kernel_launch(void* const* d_in, const int* in_sizes, int n_in,
                              void* d_out, int out_size, void* d_ws, size_t ws_size,
                              hipStream_t stream) {
  const float* x     = (const float*)d_in[0];
  const float* W1    = (const float*)d_in[1];
  const float* b1    = (const float*)d_in[2];
  const float* Wrel1 = (const float*)d_in[3];
  const float* brel1 = (const float*)d_in[4];
  const float* Wroot1= (const float*)d_in[5];
  const float* Wp    = (const float*)d_in[6];
  const float* bp    = (const float*)d_in[7];
  const float* Wrel2 = (const float*)d_in[8];
  const float* brel2 = (const float*)d_in[9];
  const float* Wroot2= (const float*)d_in[10];
  const float* W2    = (const float*)d_in[11];
  const float* b2    = (const float*)d_in[12];
  const float* W3    = (const float*)d_in[13];
  const float* b3    = (const float*)d_in[14];
  const int*   ei    = (const int*)d_in[15];
  float* ws  = (float*)d_ws;
  float* out = (float*)d_out;

  // zero atomic accumulators (agg; and as_..den contiguous tail) + loss slots
  k_zero<<<2048, 256, 0, stream>>>(ws + OFF_AGG, OFF_H1 - OFF_AGG);
  k_zero<<<2048, 256, 0, stream>>>(ws + OFF_AS, WS_END - OFF_AS);
  k_zero<<<1, 32, 0, stream>>>(out + 640, 2);

  // lin1 via WMMA: 65536/16 = 4096 waves, 4 waves/block
  k_lin1<<<1024, 128, 0, stream>>>(x, W1, b1, ws + OFF_H0);

  // edge aggregation 1 (+degrees)
  k_edge1<<<(NE * 8) / 256, 256, 0, stream>>>(ei, ws + OFF_H0, ws + OFF_AGG, ws + OFF_DEG);

  // per-node conv1 + pool logits + softmax + mincut denominator
  k_node<<<TOTAL / 256, 256, 0, stream>>>(ws + OFF_AGG, ws + OFF_H0, Wrel1, brel1,
                                          Wroot1, Wp, bp, ws + OFF_DEG,
                                          ws + OFF_H1, ws + OFF_S, ws + OFF_DEN);

  // edge aggregation 2 (A @ S)
  k_edge2<<<(NE * 4) / 256, 256, 0, stream>>>(ei, ws + OFF_S, ws + OFF_AS);

  // per-graph pooled GEMMs via WMMA: 64 graphs * 4 waves = 256 waves
  k_pool<<<64, 128, 0, stream>>>(ws + OFF_S, ws + OFF_H1, ws + OFF_AS,
                                 ws + OFF_OUT, ws + OFF_OA, ws + OFF_SS);

  // per-graph tail: losses, normalization, conv2, readout, MLP, log_softmax
  k_final<<<NGRAPH, 64, 0, stream>>>(ws + OFF_OUT, ws + OFF_OA, ws + OFF_SS,
                                     ws + OFF_DEN, Wrel2, brel2, Wroot2,
                                     W2, b2, W3, b3, out);
}